// Attention_85581518340337
// MI455X (gfx1250) — compile-verified
//
#include <hip/hip_runtime.h>
#include <hip/hip_bf16.h>
#include <math.h>

// ---------------------------------------------------------------------------
// Problem constants (from reference): B=1, DIM=64, HEADS=4, T=16, H=64, W=64
// ---------------------------------------------------------------------------
#define NTOT   65536          // T*H*W
#define C3H    768            // 3*HEADS*DIM
#define CDIM   64
#define NHEADS 4
#define TT     16

typedef __attribute__((ext_vector_type(2))) float v2f;
typedef __attribute__((ext_vector_type(8))) float v8f;

// CDNA5 fp32 matrix core: D(16x16,f32) = A(16x4,f32) * B(4x16,f32) + C
// 8-arg form: (neg_a, A, neg_b, B, c_mod, C, reuse_a, reuse_b)
__device__ __forceinline__ v8f wmma_f32_4(v2f a, v2f b, v8f c) {
    return __builtin_amdgcn_wmma_f32_16x16x4_f32(
        false, a, false, b, (short)0, c, false, false);
}

// ---------------------------------------------------------------------------
// Batched row-major GEMM, register-blocked 4x along M:
//   C[b] (MxN) = A[b] (MxK) * B[b] (KxN)
// One wave owns a 64(M) x 16(N) strip: 4 x v8f accumulators; the streamed
// B tile is loaded ONCE per 4 WMMAs. All 5 loads of a k-step are issued
// before the WMMA group so the backend can clause them and drain loadcnt
// progressively instead of stalling to zero per WMMA.
// M multiple of 64, N multiple of 16, K multiple of 16.
// A-layout per ISA 7.12.2: lane(l&15)=M row, vgprs hold K = 2*(l>>4)+{0,1}.
// B-layout mirrored: vgprs hold rows K = 2*(l>>4)+{0,1}, lane(l&15)=N col.
// ---------------------------------------------------------------------------
#define MBLK 4
__global__ __launch_bounds__(256)
void wmma_gemm(const float* __restrict__ A, const float* __restrict__ B,
               float* __restrict__ C, int M, int N, int K,
               size_t sA, size_t sB, size_t sC) {
    const int bat = blockIdx.y;
    A += (size_t)bat * sA; B += (size_t)bat * sB; C += (size_t)bat * sC;

    const int wave = (blockIdx.x * blockDim.x + threadIdx.x) >> 5;
    const int lane = threadIdx.x & 31;
    const int mgroups = M >> 6;                       // M / (16*MBLK)
    const int mg = wave % mgroups;
    const int nt = wave / mgroups;
    const int m0 = mg << 6, n0 = nt << 4;
    const int half = lane >> 4, lid = lane & 15;

    const float* Ap = A + (size_t)(m0 + lid) * K + 2 * half;   // row strip, walk K
    const float* Bp = B + (size_t)(2 * half) * N + n0 + lid;   // walk down K rows
    const size_t aRow16 = (size_t)16 * K;                      // 16 A-rows
    const size_t bStep  = (size_t)4 * N;                       // 4 B-rows / k-step

    v8f acc[MBLK] = {};
    for (int k0 = 0; k0 < K; k0 += 16) {
        // near prefetch (WGP scope) of the B stream one 16-row k-block ahead:
        // the tile is re-read by N-adjacent waves on this WGP shortly after.
        __builtin_prefetch(Bp + (size_t)16 * N, 0, 3);
        #pragma unroll
        for (int kk = 0; kk < 16; kk += 4) {
            // issue ALL loads of this k-step first (1 B pair + MBLK A pairs)
            v2f b, a[MBLK];
            b.x = Bp[0];  b.y = Bp[N];                // B[k+2h][n0+lid], B[k+2h+1][..]
            #pragma unroll
            for (int i = 0; i < MBLK; ++i) {
                a[i].x = Ap[i * aRow16];              // A[m0+16i+lid][k+2h]
                a[i].y = Ap[i * aRow16 + 1];          // A[m0+16i+lid][k+2h+1]
            }
            #pragma unroll
            for (int i = 0; i < MBLK; ++i)
                acc[i] = wmma_f32_4(a[i], b, acc[i]);
            Ap += 4;
            Bp += bStep;
        }
    }
    // C vgpr r -> row m0 + 16i + r + 8*half, col n0 + lid
    #pragma unroll
    for (int i = 0; i < MBLK; ++i) {
        float* Cp = C + (size_t)(m0 + 16 * i + 8 * half) * N + n0 + lid;
        #pragma unroll
        for (int r = 0; r < 8; ++r) Cp[(size_t)r * N] = acc[i][r];
    }
}

// ---------------------------------------------------------------------------
// Grouped 3x3x3 conv, groups = 192, 4 in-ch and 4 out-ch per group, pad=1.
// One block per (oc, t) slab; 108 weights staged in LDS; 16 outputs/thread.
// ---------------------------------------------------------------------------
__global__ __launch_bounds__(256)
void dwconv3d(const float* __restrict__ in, const float* __restrict__ w,
              float* __restrict__ out) {
    __shared__ float ws[108];
    const int oc = blockIdx.x;          // 0..767
    const int t  = blockIdx.y;          // 0..15
    const int g  = oc >> 2;             // group -> in-channel base g*4
    if (threadIdx.x < 108) ws[threadIdx.x] = w[(size_t)oc * 108 + threadIdx.x];
    __syncthreads();

    const float* inb = in + (size_t)(g * 4) * NTOT;
    float* outb = out + (size_t)oc * NTOT + (size_t)t * 4096;

    for (int p = threadIdx.x; p < 4096; p += 256) {
        const int h = p >> 6, wx = p & 63;
        float acc = 0.f;
        #pragma unroll
        for (int kd = 0; kd < 3; ++kd) {
            const int tt = t + kd - 1;
            if (tt < 0 || tt >= TT) continue;
            #pragma unroll
            for (int kh = 0; kh < 3; ++kh) {
                const int hh = h + kh - 1;
                if (hh < 0 || hh >= 64) continue;
                #pragma unroll
                for (int kw = 0; kw < 3; ++kw) {
                    const int wv = wx + kw - 1;
                    if (wv < 0 || wv >= 64) continue;
                    const size_t off = (size_t)tt * 4096 + hh * 64 + wv;
                    const int wi = kd * 9 + kh * 3 + kw;
                    #pragma unroll
                    for (int ic = 0; ic < 4; ++ic)
                        acc = fmaf(ws[ic * 27 + wi], inb[(size_t)ic * NTOT + off], acc);
                }
            }
        }
        outb[p] = acc;
    }
}

// ---------------------------------------------------------------------------
// Inverse L2 norms of rows 0..511 of qkv2 (q rows 0..255, k rows 256..511).
// ---------------------------------------------------------------------------
__global__ __launch_bounds__(256)
void rownorm(const float* __restrict__ in, float* __restrict__ invn) {
    __shared__ float red[256];
    const int row = blockIdx.x;
    const float* p = in + (size_t)row * NTOT;
    float s = 0.f;
    for (int i = threadIdx.x; i < NTOT; i += 256) { float v = p[i]; s = fmaf(v, v, s); }
    red[threadIdx.x] = s;
    __syncthreads();
    for (int st = 128; st > 0; st >>= 1) {
        if (threadIdx.x < st) red[threadIdx.x] += red[threadIdx.x + st];
        __syncthreads();
    }
    if (threadIdx.x == 0)
        invn[row] = 1.0f / fmaxf(sqrtf(red[0]), 1e-12f);
}

__global__ void zerobuf(float* __restrict__ p, int n) {
    int i = blockIdx.x * blockDim.x + threadIdx.x;
    if (i < n) p[i] = 0.f;
}

// ---------------------------------------------------------------------------
// attn_raw[h][c][d] = sum_n q[h][c][n] * k[h][d][n]  (split-K WMMA + atomics)
// 512 waves: head(4) x mg(2) x ng(2) x chunk(32 of 2048 n each).
// Each wave owns a 32x32 tile (2x2 WMMA tiles): 4 b64 loads feed 4 WMMAs.
// Both operands are read contiguously along K (the n axis).
// ---------------------------------------------------------------------------
#define QK_CHUNK 2048
__global__ __launch_bounds__(256)
void qk_gemm(const float* __restrict__ q, const float* __restrict__ k,
             float* __restrict__ attn) {
    const int wave  = (blockIdx.x * blockDim.x + threadIdx.x) >> 5;
    const int lane  = threadIdx.x & 31;
    const int chunk =  wave       & 31;
    const int ng    = (wave >> 5) & 1;
    const int mg    = (wave >> 6) & 1;
    const int h     = (wave >> 7) & 3;
    const int half = lane >> 4, lid = lane & 15;
    const int m0 = mg << 5, n0 = ng << 5;

    const float* Ap = q + (size_t)(h * 64 + m0 + lid) * NTOT + chunk * QK_CHUNK + 2 * half;
    const float* Bp = k + (size_t)(h * 64 + n0 + lid) * NTOT + chunk * QK_CHUNK + 2 * half;
    const size_t row16 = (size_t)16 * NTOT;

    v8f acc[2][2] = {};
    #pragma unroll 2
    for (int kk = 0; kk < QK_CHUNK; kk += 4) {
        v2f a0, a1, b0, b1;
        a0.x = Ap[0];         a0.y = Ap[1];
        a1.x = Ap[row16];     a1.y = Ap[row16 + 1];
        b0.x = Bp[0];         b0.y = Bp[1];
        b1.x = Bp[row16];     b1.y = Bp[row16 + 1];
        acc[0][0] = wmma_f32_4(a0, b0, acc[0][0]);
        acc[0][1] = wmma_f32_4(a0, b1, acc[0][1]);
        acc[1][0] = wmma_f32_4(a1, b0, acc[1][0]);
        acc[1][1] = wmma_f32_4(a1, b1, acc[1][1]);
        Ap += 4; Bp += 4;
    }
    #pragma unroll
    for (int i = 0; i < 2; ++i) {
        #pragma unroll
        for (int j = 0; j < 2; ++j) {
            float* dst = attn + (size_t)h * 4096
                       + (size_t)(m0 + 16 * i + 8 * half) * 64 + n0 + 16 * j + lid;
            #pragma unroll
            for (int r = 0; r < 8; ++r) atomicAdd(dst + (size_t)r * 64, acc[i][j][r]);
        }
    }
}

// ---------------------------------------------------------------------------
// Row softmax over d with fused scaling:  x = raw * invq[c] * invk[d] * temp[h].
// One wave32 per (h, c) row of 64; 2 elems/lane; shfl_xor reductions.
// ---------------------------------------------------------------------------
__global__ __launch_bounds__(256)
void softmax_attn(float* __restrict__ attn, const float* __restrict__ invn,
                  const float* __restrict__ temp) {
    const int wave = (blockIdx.x * blockDim.x + threadIdx.x) >> 5;
    const int lane = threadIdx.x & 31;
    const int h = wave >> 6;         // 0..3
    const int c = wave & 63;         // 0..63
    float* row = attn + (size_t)h * 4096 + (size_t)c * 64;
    const float iq = invn[h * 64 + c];
    const float tp = temp[h];

    float x0 = row[lane]      * iq * invn[256 + h * 64 + lane]      * tp;
    float x1 = row[lane + 32] * iq * invn[256 + h * 64 + lane + 32] * tp;
    float m = fmaxf(x0, x1);
    #pragma unroll
    for (int s = 16; s > 0; s >>= 1) m = fmaxf(m, __shfl_xor(m, s, 32));
    const float e0 = expf(x0 - m), e1 = expf(x1 - m);
    float sum = e0 + e1;
    #pragma unroll
    for (int s = 16; s > 0; s >>= 1) sum += __shfl_xor(sum, s, 32);
    const float inv = 1.0f / sum;
    row[lane]      = e0 * inv;
    row[lane + 32] = e1 * inv;
}

// ---------------------------------------------------------------------------
// Launch: x, qkv_w, dw_w, proj_w, temperature  ->  out (1,64,16,64,64) f32
// Scratch layout (floats): [ qkv 768*65536 | dw 768*65536 | invn 512 | attn 16384 ]
// The first 192 MiB buffer is reused for the attention output (256*65536).
// ---------------------------------------------------------------------------
extern "C" void kernel_launch(void* const* d_in, const int* in_sizes, int n_in,
                              void* d_out, int out_size, void* d_ws, size_t ws_size,
                              hipStream_t stream) {
    (void)in_sizes; (void)n_in; (void)out_size; (void)ws_size;
    const float* x      = (const float*)d_in[0];   // (64, 65536)
    const float* qkv_w  = (const float*)d_in[1];   // (768, 64)
    const float* dw_w   = (const float*)d_in[2];   // (768, 108)
    const float* proj_w = (const float*)d_in[3];   // (64, 256)
    const float* temp   = (const float*)d_in[4];   // (4)
    float* out = (float*)d_out;

    float* buf_qkv = (float*)d_ws;                              // 768*65536
    float* buf_dw  = buf_qkv + (size_t)C3H * NTOT;              // 768*65536
    float* inv_n   = buf_dw  + (size_t)C3H * NTOT;              // 512
    float* attn    = inv_n + 512;                               // 4*64*64
    float* buf_out = buf_qkv;                                   // reuse: 256*65536

    // 1) qkv = W_qkv (768x64) @ x (64x65536)     [12 x 4096 wave strips]
    {
        const int waves = (C3H / 64) * (NTOT / 16);             // 49152
        wmma_gemm<<<dim3(waves / 8, 1), 256, 0, stream>>>(
            qkv_w, x, buf_qkv, C3H, NTOT, CDIM, 0, 0, 0);
    }
    // 2) grouped 3x3x3 conv (pad 1)
    dwconv3d<<<dim3(C3H, TT), 256, 0, stream>>>(buf_qkv, dw_w, buf_dw);

    // 3) inverse L2 norms of the 512 q/k rows
    rownorm<<<512, 256, 0, stream>>>(buf_dw, inv_n);

    // 4) zero attn accumulator (required every call: split-K uses atomics)
    zerobuf<<<64, 256, 0, stream>>>(attn, NHEADS * 64 * 64);

    // 5) attn_raw = q @ k^T  (split-K 32 x 2048, 2x2 blocked, f32 atomics)
    qk_gemm<<<64, 256, 0, stream>>>(buf_dw, buf_dw + (size_t)256 * NTOT, attn);

    // 6) fused scale (1/|q| * 1/|k| * temperature) + softmax over d
    softmax_attn<<<32, 256, 0, stream>>>(attn, inv_n, temp);

    // 7) out_h = attn_h (64x64) @ v_h (64x65536), batched over 4 heads
    //    one wave covers the full M=64 extent: v streamed exactly once.
    {
        const int waves = (CDIM / 64) * (NTOT / 16);            // 4096 per head
        wmma_gemm<<<dim3(waves / 8, NHEADS), 256, 0, stream>>>(
            attn, buf_dw + (size_t)512 * NTOT, buf_out,
            CDIM, NTOT, CDIM,
            (size_t)64 * 64, (size_t)64 * NTOT, (size_t)64 * NTOT);
    }
    // 8) final = W_proj (64x256) @ out (256x65536)
    {
        const int waves = (CDIM / 64) * (NTOT / 16);
        wmma_gemm<<<dim3(waves / 8, 1), 256, 0, stream>>>(
            proj_w, buf_out, out, CDIM, NTOT, 4 * CDIM, 0, 0, 0);
    }
}